// ScratchMPNN_72232759984910
// MI455X (gfx1250) — compile-verified
//
#include <hip/hip_runtime.h>

typedef __attribute__((ext_vector_type(2))) float v2f;
typedef __attribute__((ext_vector_type(8))) float v8f;

#define HIDC 128
#define NODES_PER_BLOCK 128
#define N_CLASSES 10
#define NUM_GRAPHS 64

// ---------------------------------------------------------------------------
// Y[M x 128] = relu?( concat(A0,A1)[M x K] @ W[128 x K]^T + bias )
// K = 128 (A1 unused) or K = 256 (cols 0..127 from A0, 128..255 from A1).
// One block = 8 wave32 = 128 node rows; each wave owns a 16x128 strip.
// W K-slice (128x128 f32 = 64KB) staged in LDS; fp32 WMMA 16x16x4.
// ---------------------------------------------------------------------------
__global__ __launch_bounds__(256)
void mpnn_gemm_wmma(const float* __restrict__ A0, const float* __restrict__ A1,
                    const float* __restrict__ W, const float* __restrict__ bias,
                    float* __restrict__ Y, int M, int K, int doRelu)
{
    __shared__ float sW[HIDC * 128]; // 64 KB
    const int tid   = threadIdx.x;
    const int wave  = tid >> 5;
    const int lane  = tid & 31;
    const int laneN = lane & 15;
    const int hi    = lane >> 4;               // which half-wave (K/M split)
    const int m0    = blockIdx.x * NODES_PER_BLOCK + wave * 16;

    v8f acc[8];
#pragma unroll
    for (int t = 0; t < 8; ++t) {
        float bv = bias[t * 16 + laneN];       // D element (r,lane) is column n0+laneN
        v8f b8 = {bv, bv, bv, bv, bv, bv, bv, bv};
        acc[t] = b8;
    }

    const int nSlices = K >> 7;                // 1 or 2
    for (int s = 0; s < nSlices; ++s) {
        const float* __restrict__ Aptr = (s == 0) ? A0 : A1;
        const int kbase = s << 7;

        // Cooperative LDS fill: sW[n][kk] = W[n][kbase+kk], 4096 float4s / 256 thr
#pragma unroll
        for (int i = 0; i < 16; ++i) {
            int v   = tid + i * 256;           // float4 index in slice
            int n   = v >> 5;                  // 32 float4 per 128-col row
            int kk4 = v & 31;
            float4 val = ((const float4*)(W + (long long)n * K + kbase))[kk4];
            ((float4*)sW)[n * 32 + kk4] = val;
        }
        __syncthreads();

        if (m0 < M) {                          // whole wave uniform -> EXEC all 1s
            const float* aRow = Aptr + (long long)(m0 + laneN) * HIDC;
            for (int kk = 0; kk < 128; kk += 4) {
                // A frag: lane holds row m=laneN; VGPR v = K (kk + 2*hi + v)
                v2f a = *(const v2f*)(aRow + kk + hi * 2);
#pragma unroll
                for (int t = 0; t < 8; ++t) {
                    // B frag: lane holds col n = 16t+laneN; B[k][n] = W[n][k]
                    v2f b = *(const v2f*)(&sW[(t * 16 + laneN) * 128 + kk + hi * 2]);
                    acc[t] = __builtin_amdgcn_wmma_f32_16x16x4_f32(
                        false, a, false, b, (short)0, acc[t], false, false);
                }
            }
        }
        __syncthreads();
    }

    if (m0 < M) {
#pragma unroll
        for (int t = 0; t < 8; ++t) {
#pragma unroll
            for (int r = 0; r < 8; ++r) {
                float v = acc[t][r];           // row m0 + r + 8*hi, col 16t+laneN
                if (doRelu) v = fmaxf(v, 0.0f);
                Y[(long long)(m0 + r + hi * 8) * HIDC + t * 16 + laneN] = v;
            }
        }
    }
}

// ---------------------------------------------------------------------------
// aggr[row[e]] += Y[col[e]]  — one wave per edge, float4 gather + f32 atomics
// ---------------------------------------------------------------------------
__global__ __launch_bounds__(256)
void scatter_add_edges(const float* __restrict__ Y, const int* __restrict__ row,
                       const int* __restrict__ col, float* __restrict__ aggr, int E)
{
    int e = blockIdx.x * 8 + (threadIdx.x >> 5);
    if (e >= E) return;
    int lane = threadIdx.x & 31;
    int r = row[e];
    int c = col[e];
    float4 v = ((const float4*)(Y + (long long)c * HIDC))[lane];
    float* dst = aggr + (long long)r * HIDC + lane * 4;
    unsafeAtomicAdd(dst + 0, v.x);
    unsafeAtomicAdd(dst + 1, v.y);
    unsafeAtomicAdd(dst + 2, v.z);
    unsafeAtomicAdd(dst + 3, v.w);
}

// ---------------------------------------------------------------------------
// pooled[batch[i]] += h[i]; cnt[batch[i]] += 1 — one wave per node
// ---------------------------------------------------------------------------
__global__ __launch_bounds__(256)
void pool_nodes(const float* __restrict__ h, const int* __restrict__ batch,
                float* __restrict__ pooled, float* __restrict__ cnt, int N)
{
    int node = blockIdx.x * 8 + (threadIdx.x >> 5);
    if (node >= N) return;
    int lane = threadIdx.x & 31;
    int g = batch[node];
    float4 v = ((const float4*)(h + (long long)node * HIDC))[lane];
    float* dst = pooled + g * HIDC + lane * 4;
    unsafeAtomicAdd(dst + 0, v.x);
    unsafeAtomicAdd(dst + 1, v.y);
    unsafeAtomicAdd(dst + 2, v.z);
    unsafeAtomicAdd(dst + 3, v.w);
    if (lane == 0) unsafeAtomicAdd(cnt + g, 1.0f);
}

// out[g][k] = (pooled[g]/cnt[g]) . W_cls[k] + b_cls[k]   (64x10, trivial)
__global__ void classify(const float* __restrict__ pooled, const float* __restrict__ cnt,
                         const float* __restrict__ Wc, const float* __restrict__ bc,
                         float* __restrict__ out)
{
    int i = blockIdx.x * blockDim.x + threadIdx.x;
    if (i >= NUM_GRAPHS * N_CLASSES) return;
    int g = i / N_CLASSES, k = i % N_CLASSES;
    float inv = 1.0f / cnt[g];
    float s = bc[k];
    for (int c = 0; c < HIDC; ++c)
        s += pooled[g * HIDC + c] * inv * Wc[k * HIDC + c];
    out[i] = s;
}

__global__ void zero_f32(float* __restrict__ p, long long n)
{
    long long i = (long long)blockIdx.x * blockDim.x + threadIdx.x;
    long long stride = (long long)gridDim.x * blockDim.x;
    for (; i < n; i += stride) p[i] = 0.0f;
}

extern "C" void kernel_launch(void* const* d_in, const int* in_sizes, int n_in,
                              void* d_out, int out_size, void* d_ws, size_t ws_size,
                              hipStream_t stream) {
    const float* x       = (const float*)d_in[0];
    const int*   edges   = (const int*)d_in[1];   // [2, E] flat: rows then cols
    const int*   batch   = (const int*)d_in[2];
    const float* W_msg1  = (const float*)d_in[3];
    const float* b_msg1  = (const float*)d_in[4];
    const float* W_upd1  = (const float*)d_in[5];
    const float* b_upd1  = (const float*)d_in[6];
    const float* W_msg2  = (const float*)d_in[7];
    const float* b_msg2  = (const float*)d_in[8];
    const float* W_upd2  = (const float*)d_in[9];
    const float* b_upd2  = (const float*)d_in[10];
    const float* W_cls   = (const float*)d_in[11];
    const float* b_cls   = (const float*)d_in[12];
    float* out = (float*)d_out;

    const int N = in_sizes[0] / HIDC;          // 50000
    const int E = in_sizes[1] / 2;             // 800000
    const int* row = edges;
    const int* col = edges + E;

    float* bufY    = (float*)d_ws;             // N*128
    float* bufAggr = bufY    + (long long)N * HIDC;
    float* bufH    = bufAggr + (long long)N * HIDC;
    float* pooled  = bufH    + (long long)N * HIDC;
    float* cnt     = pooled + NUM_GRAPHS * HIDC;

    const int gemmGrid    = (N + NODES_PER_BLOCK - 1) / NODES_PER_BLOCK;
    const int edgeGrid    = (E + 7) / 8;
    const int nodeGrid    = (N + 7) / 8;
    const long long nFeat = (long long)N * HIDC;

    // ---- Layer 1 ----
    // Y = x @ W_msg1^T + b_msg1 (per-node; message depends only on source node)
    mpnn_gemm_wmma<<<gemmGrid, 256, 0, stream>>>(x, nullptr, W_msg1, b_msg1, bufY, N, 128, 0);
    zero_f32<<<2048, 256, 0, stream>>>(bufAggr, nFeat);
    scatter_add_edges<<<edgeGrid, 256, 0, stream>>>(bufY, row, col, bufAggr, E);
    // h1 = relu([x, aggr] @ W_upd1^T + b_upd1)
    mpnn_gemm_wmma<<<gemmGrid, 256, 0, stream>>>(x, bufAggr, W_upd1, b_upd1, bufH, N, 256, 1);

    // ---- Layer 2 ----
    mpnn_gemm_wmma<<<gemmGrid, 256, 0, stream>>>(bufH, nullptr, W_msg2, b_msg2, bufY, N, 128, 0);
    zero_f32<<<2048, 256, 0, stream>>>(bufAggr, nFeat);
    scatter_add_edges<<<edgeGrid, 256, 0, stream>>>(bufY, row, col, bufAggr, E);
    // h2 = relu([h1, aggr] @ W_upd2^T + b_upd2)  (overwrite bufY; h1 stays in bufH)
    mpnn_gemm_wmma<<<gemmGrid, 256, 0, stream>>>(bufH, bufAggr, W_upd2, b_upd2, bufY, N, 256, 1);

    // ---- Pool + classify ----
    zero_f32<<<64, 256, 0, stream>>>(pooled, NUM_GRAPHS * HIDC + NUM_GRAPHS);
    pool_nodes<<<nodeGrid, 256, 0, stream>>>(bufY, batch, pooled, cnt, N);
    classify<<<10, 64, 0, stream>>>(pooled, cnt, W_cls, b_cls, out);
}